// train_time_calibration_56461640073699
// MI455X (gfx1250) — compile-verified
//
#include <hip/hip_runtime.h>
#include <hip/hip_bf16.h>

// Depthwise 51-tap convolution along H as banded-Toeplitz GEMM on the CDNA5
// f32 WMMA unit: out(16x16 tile) = A(16x68) * Xpatch(68x16), accumulated as
// 17 x V_WMMA_F32_16X16X4_F32 split over two accumulators.
//
// v3: ping-pong double buffering of B operands (loads for tile i+1 issue
// before the WMMA chain of tile i), 4-wave workgroups (one h-tile row per
// wave, readfirstlane-forced uniform so interior/boundary stays a scalar
// branch and WMMA always runs with EXEC all-ones).

typedef __attribute__((ext_vector_type(2))) float v2f;
typedef __attribute__((ext_vector_type(8))) float v8f;

#define BB     64
#define HH     2048
#define WW     1024
#define NDEPTH 9
#define FLEN   51
#define PAD    25
#define NCHUNK 17   // ceil((16 + 51 - 1) / 4) = ceil(66/4)
#define WT     8    // 16-wide W tiles per wave
#define WAVES  4    // waves per workgroup (each takes its own h-tile)

__device__ __forceinline__ void load_tile(const float* __restrict__ p,
                                          float* __restrict__ Bv)
{
#pragma unroll
    for (int c = 0; c < NCHUNK; ++c) {
        Bv[2 * c + 0] = p[(size_t)(4 * c)     * WW];
        Bv[2 * c + 1] = p[(size_t)(4 * c + 1) * WW];
    }
}

__device__ __forceinline__ void compute_store(const float* __restrict__ A0,
                                              const float* __restrict__ A1,
                                              const float* __restrict__ Bv,
                                              float* __restrict__ q)
{
    v8f acc0 = {};
    v8f acc1 = {};
#pragma unroll
    for (int c = 0; c < NCHUNK; c += 2) {
        v2f a  = { A0[c], A1[c] };
        v2f bv = { Bv[2 * c], Bv[2 * c + 1] };
        acc0 = __builtin_amdgcn_wmma_f32_16x16x4_f32(
                   false, a, false, bv, (short)0, acc0, false, false);
    }
#pragma unroll
    for (int c = 1; c < NCHUNK; c += 2) {
        v2f a  = { A0[c], A1[c] };
        v2f bv = { Bv[2 * c], Bv[2 * c + 1] };
        acc1 = __builtin_amdgcn_wmma_f32_16x16x4_f32(
                   false, a, false, bv, (short)0, acc1, false, false);
    }
#pragma unroll
    for (int r = 0; r < 8; ++r) {
        q[(size_t)r * WW] = acc0[r] + acc1[r];
    }
}

__global__ __launch_bounds__(WAVES * 32)
void dwconv_h_wmma(const float* __restrict__ x,
                   const float* __restrict__ kernels,
                   const int*   __restrict__ depth,
                   float*       __restrict__ out)
{
    const int lane = (int)threadIdx.x & 31;  // wave32 lane
    const int m    = lane & 15;              // matrix row (A) / column (B, D)
    const int half = lane >> 4;              // 0: lanes 0-15, 1: lanes 16-31
    const int koff = half * 2;               // K offset of this half-wave's VGPR0

    // Force wave-uniformity so h0-derived control flow is a scalar branch
    // (WMMA must never sit under a divergent exec mask).
    const int waveId = __builtin_amdgcn_readfirstlane((int)threadIdx.x >> 5);

    const int b      = (int)blockIdx.z;
    const int h0     = ((int)blockIdx.y * WAVES + waveId) * 16;
    const int w0base = (int)blockIdx.x * (WT * 16);

    const float* taps = kernels + (size_t)depth[b] * FLEN;

    // A-operand registers, built once per wave.
    // A[m, j] = taps[j - m] for 0 <= j - m < FLEN, else 0.
    // Layout: a[0] = A[m, 4c + koff], a[1] = A[m, 4c + koff + 1].
    float A0[NCHUNK], A1[NCHUNK];
#pragma unroll
    for (int c = 0; c < NCHUNK; ++c) {
        const int j0 = 4 * c + koff - m;
        const int j1 = j0 + 1;
        A0[c] = (j0 >= 0 && j0 < FLEN) ? taps[j0] : 0.0f;
        A1[c] = (j1 >= 0 && j1 < FLEN) ? taps[j1] : 0.0f;
    }

    const bool   interior  = (h0 >= PAD) && (h0 - PAD + 4 * (NCHUNK - 1) + 3 < HH);
    const size_t planeBase = (size_t)b * HH * WW;

    float* qbase = out + planeBase + (size_t)(h0 + 8 * half) * WW + w0base + m;

    if (interior) {
        const float* pbase = x + planeBase
                             + (size_t)(h0 - PAD + koff) * WW + w0base + m;
        float Bping[2 * NCHUNK], Bpong[2 * NCHUNK];

        load_tile(pbase, Bping);                         // tile 0
#pragma unroll
        for (int wt = 0; wt < WT; wt += 2) {
            load_tile(pbase + (wt + 1) * 16, Bpong);     // prefetch tile wt+1
            compute_store(A0, A1, Bping, qbase + wt * 16);
            if (wt + 2 < WT)
                load_tile(pbase + (wt + 2) * 16, Bping); // prefetch tile wt+2
            compute_store(A0, A1, Bpong, qbase + (wt + 1) * 16);
        }
    } else {
        // Boundary h-tiles (4 of 128): clamp row index (always-legal address)
        // and zero out-of-range rows via mask multiply. Not pipelined.
        const float* pb = x + planeBase + w0base + m;
        for (int wt = 0; wt < WT; ++wt) {
            float Bv[2 * NCHUNK];
#pragma unroll
            for (int c = 0; c < NCHUNK; ++c) {
                const int hin0 = h0 - PAD + 4 * c + koff;
                const int hin1 = hin0 + 1;
                const int hc0  = hin0 < 0 ? 0 : (hin0 >= HH ? HH - 1 : hin0);
                const int hc1  = hin1 < 0 ? 0 : (hin1 >= HH ? HH - 1 : hin1);
                const float m0 = (hin0 == hc0) ? 1.0f : 0.0f;
                const float m1 = (hin1 == hc1) ? 1.0f : 0.0f;
                Bv[2 * c + 0] = pb[(size_t)hc0 * WW + wt * 16] * m0;
                Bv[2 * c + 1] = pb[(size_t)hc1 * WW + wt * 16] * m1;
            }
            compute_store(A0, A1, Bv, qbase + wt * 16);
        }
    }
}

extern "C" void kernel_launch(void* const* d_in, const int* in_sizes, int n_in,
                              void* d_out, int out_size, void* d_ws, size_t ws_size,
                              hipStream_t stream)
{
    const float* x       = (const float*)d_in[0];
    const float* kernels = (const float*)d_in[1];
    const int*   depth   = (const int*)d_in[2];
    float*       out     = (float*)d_out;

    dim3 grid(WW / (16 * WT), (HH / 16) / WAVES, BB);  // (8, 32, 64)
    dim3 block(WAVES * 32, 1, 1);                      // 4 waves per workgroup
    dwconv_h_wmma<<<grid, block, 0, stream>>>(x, kernels, depth, out);
}